// Group_83588653515193
// MI455X (gfx1250) — compile-verified
//
#include <hip/hip_runtime.h>

// Problem constants (from setup_inputs): B=32, N=8192, G=512, K=32
#define BATCH 32
#define NPTS  8192
#define NGRP  512
#define KNN   32

typedef __attribute__((ext_vector_type(2))) float v2f;
typedef __attribute__((ext_vector_type(8))) float v8f;

// ---------------------------------------------------------------------------
// Kernel 1: farthest point sampling. One workgroup per batch.
// Writes center coords [B, G, 3] into centerOut.
// ---------------------------------------------------------------------------
#define FPS_BLK 256

__global__ __launch_bounds__(FPS_BLK)
void fps_kernel(const float* __restrict__ xyz, float* __restrict__ centerOut) {
    __shared__ float mind[NPTS];          // 32 KB
    __shared__ float redV[FPS_BLK];
    __shared__ int   redI[FPS_BLK];

    const int b   = blockIdx.x;
    const int tid = threadIdx.x;
    const float* p = xyz + (size_t)b * NPTS * 3;
    float* cout = centerOut + (size_t)b * NGRP * 3;

    // first sample = point 0
    float lx = p[0], ly = p[1], lz = p[2];
    if (tid == 0) { cout[0] = lx; cout[1] = ly; cout[2] = lz; }
    for (int i = tid; i < NPTS; i += FPS_BLK) mind[i] = 1e10f;
    __syncthreads();

    for (int s = 1; s < NGRP; ++s) {
        float bv = -1.0f; int bi = 0;
        for (int i = tid; i < NPTS; i += FPS_BLK) {
            float dx = p[3 * i + 0] - lx;
            float dy = p[3 * i + 1] - ly;
            float dz = p[3 * i + 2] - lz;
            float d  = dx * dx + dy * dy + dz * dz;
            float m  = fminf(mind[i], d);
            mind[i]  = m;
            if (m > bv) { bv = m; bi = i; }   // strict > keeps lowest index on ties
        }
        redV[tid] = bv; redI[tid] = bi;
        __syncthreads();
        for (int off = FPS_BLK >> 1; off > 0; off >>= 1) {
            if (tid < off) {
                float ov = redV[tid + off]; int oi = redI[tid + off];
                if (ov > redV[tid] || (ov == redV[tid] && oi < redI[tid])) {
                    redV[tid] = ov; redI[tid] = oi;
                }
            }
            __syncthreads();
        }
        int best = redI[0];
        lx = p[3 * best + 0]; ly = p[3 * best + 1]; lz = p[3 * best + 2];
        if (tid == 0) { cout[3 * s + 0] = lx; cout[3 * s + 1] = ly; cout[3 * s + 2] = lz; }
        __syncthreads();   // protect redV/redI before next iteration's writes
    }
}

// ---------------------------------------------------------------------------
// Kernel 2: KNN via V_WMMA_F32_16X16X4_F32 + top-32 selection + gather.
// One 256-thread block per (batch, tile of 16 centers).
// Score for ordering: s = |p|^2 - 2*(c . p)   (the |c|^2 term is a per-center
// constant and cannot change the top-k ordering).
// ---------------------------------------------------------------------------
#define CHUNK 512           // points staged per LDS round
#define KNN_BLK 256         // 8 waves

__global__ __launch_bounds__(KNN_BLK)
void knn_kernel(const float* __restrict__ xyz,
                const float* __restrict__ centerAll,
                float* __restrict__ nbhOut) {
    __shared__ float cen[16 * 3];
    __shared__ float score[CHUNK * 16];   // [point_in_chunk][center]  32 KB
    __shared__ float kV[16][KNN];
    __shared__ int   kI[16][KNN];

    const int tid  = threadIdx.x;
    const int lane = tid & 31;
    const int wave = tid >> 5;            // 0..7
    const int b    = blockIdx.x >> 5;     // / (NGRP/16)
    const int gt   = blockIdx.x & 31;     // % (NGRP/16)

    const float* p = xyz + (size_t)b * NPTS * 3;
    const float* cbase = centerAll + ((size_t)b * NGRP + gt * 16) * 3;

    if (tid < 48) cen[tid] = cbase[tid];
    for (int i = tid; i < 16 * KNN; i += KNN_BLK) {
        kV[i / KNN][i % KNN] = 3.0e38f;
        kI[i / KNN][i % KNN] = 0;
    }
    __syncthreads();

    // A matrix (16 centers x 4, K padded): lanes 0-15 hold (x,y)=K0,K1 of
    // center M=lane; lanes 16-31 hold (z,0)=K2,K3 of center M=lane-16.
    const int  col = lane & 15;
    const bool lo  = lane < 16;
    v2f a;
    a.x = lo ? cen[col * 3 + 0] : cen[col * 3 + 2];
    a.y = lo ? cen[col * 3 + 1] : 0.0f;

    for (int chunk = 0; chunk < NPTS / CHUNK; ++chunk) {
        const int cb = chunk * CHUNK;
        if (chunk + 1 < NPTS / CHUNK)
            __builtin_prefetch(p + (size_t)(cb + CHUNK) * 3 + (size_t)tid * 24, 0, 1);

        // 32 tiles of 16 points per chunk; 4 tiles per wave (uniform -> EXEC full)
        for (int t = wave; t < CHUNK / 16; t += KNN_BLK / 32) {
            const int pt = cb + t * 16 + col;     // this lane's column point
            // B matrix (4 x 16 points): same half-lane split as A
            float b0 = p[3 * pt + (lo ? 0 : 2)];
            float b1 = lo ? p[3 * pt + 1] : 0.0f;
            v2f bb; bb.x = b0; bb.y = b1;

            // |p|^2 for this column: combine the two half-lane partials
            float partial = b0 * b0 + b1 * b1;
            float pn2 = __shfl(partial, col) + __shfl(partial, col + 16);

            v8f acc = {};
            v8f d = __builtin_amdgcn_wmma_f32_16x16x4_f32(
                false, a, false, bb, (short)0, acc, false, false);

            // D layout: VGPR r, this lane -> center M = r + (lo?0:8), point = col
            float s[8];
#pragma unroll
            for (int r = 0; r < 8; ++r) s[r] = fmaf(-2.0f, d[r], pn2);

            const int moff = lo ? 0 : 8;
            float* dst = &score[(t * 16 + col) * 16 + moff];
            ((float4*)dst)[0] = make_float4(s[0], s[1], s[2], s[3]);
            ((float4*)dst)[1] = make_float4(s[4], s[5], s[6], s[7]);
        }
        __syncthreads();

        // top-32 sorted insertion, one thread per center
        if (tid < 16) {
            float* v  = kV[tid];
            int*   ix = kI[tid];
            for (int pp = 0; pp < CHUNK; ++pp) {
                float s = score[pp * 16 + tid];
                if (s < v[KNN - 1]) {
                    int j = KNN - 1;
                    while (j > 0 && v[j - 1] > s) {
                        v[j] = v[j - 1]; ix[j] = ix[j - 1]; --j;
                    }
                    v[j] = s; ix[j] = cb + pp;
                }
            }
        }
        __syncthreads();
    }

    // gather neighborhood and subtract center
    for (int u = tid; u < 16 * KNN; u += KNN_BLK) {
        const int g = u / KNN, k = u % KNN;
        const int idx = kI[g][k];
        const float cx = cen[g * 3 + 0];
        const float cy = cen[g * 3 + 1];
        const float cz = cen[g * 3 + 2];
        size_t o = (((size_t)b * NGRP + (size_t)gt * 16 + g) * KNN + k) * 3;
        nbhOut[o + 0] = p[3 * idx + 0] - cx;
        nbhOut[o + 1] = p[3 * idx + 1] - cy;
        nbhOut[o + 2] = p[3 * idx + 2] - cz;
    }
}

// ---------------------------------------------------------------------------
extern "C" void kernel_launch(void* const* d_in, const int* in_sizes, int n_in,
                              void* d_out, int out_size, void* d_ws, size_t ws_size,
                              hipStream_t stream) {
    (void)in_sizes; (void)n_in; (void)d_ws; (void)ws_size; (void)out_size;
    const float* xyz = (const float*)d_in[0];
    float* out = (float*)d_out;

    // d_out layout: neighborhood [B,G,K,3] then center [B,G,3]
    float* nbh = out;
    float* center = out + (size_t)BATCH * NGRP * KNN * 3;

    fps_kernel<<<BATCH, FPS_BLK, 0, stream>>>(xyz, center);
    knn_kernel<<<BATCH * (NGRP / 16), KNN_BLK, 0, stream>>>(xyz, center, nbh);
}